// mlpNet_40252433498128
// MI455X (gfx1250) — compile-verified
//
#include <hip/hip_runtime.h>

typedef __attribute__((ext_vector_type(2))) float v2f;
typedef __attribute__((ext_vector_type(8))) float v8f;

#define N_V 100000
#define N_E 20000
#define N_PAIRS 1600000
#define F 128
#define H 128
#define EPS 1e-5f

// ---------------------------------------------------------------- zero fill
__global__ void zero_f32(float* __restrict__ p, long n) {
    long i = (long)blockIdx.x * blockDim.x + threadIdx.x;
    long stride = (long)gridDim.x * blockDim.x;
    for (; i < n; i += stride) p[i] = 0.0f;
}

// ------------------------------------------- GEMM h = X*W + b  (+ BN stats)
// One wave per 16x16 output tile. blockIdx.x = M-tile (6250), wave = N-tile (8).
// K-loop: 32 steps of V_WMMA_F32_16X16X4_F32.
__global__ __launch_bounds__(256)
void gemm_bn_stats(const float* __restrict__ X, const float* __restrict__ W,
                   const float* __restrict__ bias,
                   float* __restrict__ h,
                   float* __restrict__ ch_sum, float* __restrict__ ch_sumsq) {
    const int wave = threadIdx.x >> 5;
    const int lane = threadIdx.x & 31;
    const int m0 = blockIdx.x * 16;
    const int n0 = wave * 16;
    const int half = lane >> 4;   // 0: lanes 0-15, 1: lanes 16-31
    const int lm   = lane & 15;

    v8f c = {};
    const float* __restrict__ Arow = X + (size_t)(m0 + lm) * F;

    // A 16x4 fp32: lanes 0-15 hold K={k0,k0+1}, lanes 16-31 hold K={k0+2,k0+3}
    // B 4x16  fp32: VGPR0 row K=k0 (lanes 0-15) / K=k0+2 (lanes 16-31), VGPR1 +1
#pragma unroll 4
    for (int kk = 0; kk < F / 4; ++kk) {
        const int kb = kk * 4 + half * 2;
        v2f a = *(const v2f*)(Arow + kb);          // 8B aligned (kb even)
        v2f b;
        b.x = W[(size_t)kb * H + n0 + lm];
        b.y = W[(size_t)(kb + 1) * H + n0 + lm];
        c = __builtin_amdgcn_wmma_f32_16x16x4_f32(
                /*neg_a=*/false, a, /*neg_b=*/false, b,
                /*c_mod=*/(short)0, c, /*reuse_a=*/false, /*reuse_b=*/false);
    }

    // Epilogue: bias add, store h, per-channel sum / sum-of-squares.
    // C layout: VGPR i -> row m0 + half*8 + i; all 8 values in column n0+lm.
    const int n = n0 + lm;
    const float bn = bias[n];
    float s = 0.0f, s2 = 0.0f;
#pragma unroll
    for (int i = 0; i < 8; ++i) {
        float v = c[i] + bn;
        s  += v;
        s2 += v * v;
        h[(size_t)(m0 + half * 8 + i) * H + n] = v;
    }
    // fold the two half-waves (same channel n), then one atomic per channel
    s  += __shfl_xor(s, 16, 32);
    s2 += __shfl_xor(s2, 16, 32);
    if (half == 0) {
        atomicAdd(&ch_sum[n],   s);
        atomicAdd(&ch_sumsq[n], s2);
    }
}

// ---------------------------------------------- BN: per-channel scale/shift
__global__ void bn_finalize(const float* __restrict__ ch_sum,
                            const float* __restrict__ ch_sumsq,
                            const float* __restrict__ gamma,
                            const float* __restrict__ beta,
                            float* __restrict__ scale, float* __restrict__ shift) {
    int ch = threadIdx.x;
    if (ch < H) {
        float mu  = ch_sum[ch] * (1.0f / (float)N_V);
        float var = ch_sumsq[ch] * (1.0f / (float)N_V) - mu * mu;
        float rs  = rsqrtf(var + EPS);
        float sc  = rs * gamma[ch];
        scale[ch] = sc;
        shift[ch] = beta[ch] - mu * sc;
    }
}

// ------------------------- pair scatter: normalized h[v] accumulated into Xe[e]
// One wave per pair; each lane owns 4 channels (float4 = b128 load).
__global__ __launch_bounds__(256)
void scatter_v2e(const float* __restrict__ h,
                 const int* __restrict__ vid, const int* __restrict__ eid,
                 const float* __restrict__ scale, const float* __restrict__ shift,
                 float* __restrict__ Xe, float* __restrict__ ecnt) {
    const int wave = threadIdx.x >> 5;
    const int lane = threadIdx.x & 31;
    const long pair = (long)blockIdx.x * 8 + wave;
    if (pair >= N_PAIRS) return;          // wave-uniform
    const int v = vid[pair];
    const int e = eid[pair];
    const int c0 = lane * 4;
    const float4 hv = *(const float4*)(h + (size_t)v * H + c0);
    const float4 sc = *(const float4*)(scale + c0);
    const float4 sh = *(const float4*)(shift + c0);
    float* dst = Xe + (size_t)e * H + c0;
    atomicAdd(dst + 0, fmaf(hv.x, sc.x, sh.x));
    atomicAdd(dst + 1, fmaf(hv.y, sc.y, sh.y));
    atomicAdd(dst + 2, fmaf(hv.z, sc.z, sh.z));
    atomicAdd(dst + 3, fmaf(hv.w, sc.w, sh.w));
    if (lane == 0) atomicAdd(ecnt + e, 1.0f);
}

// ------------------------------------------------- divide rows by segment count
__global__ void seg_div(float* __restrict__ buf, const float* __restrict__ cnt,
                        long total) {
    long idx = (long)blockIdx.x * blockDim.x + threadIdx.x;
    if (idx < total) {
        long r = idx >> 7;                 // H == 128
        buf[idx] = buf[idx] / fmaxf(cnt[r], 1.0f);
    }
}

// ------------------------- pair scatter: Xe[e] accumulated into out[v]
__global__ __launch_bounds__(256)
void scatter_e2v(const float* __restrict__ Xe,
                 const int* __restrict__ vid, const int* __restrict__ eid,
                 float* __restrict__ out, float* __restrict__ vcnt) {
    const int wave = threadIdx.x >> 5;
    const int lane = threadIdx.x & 31;
    const long pair = (long)blockIdx.x * 8 + wave;
    if (pair >= N_PAIRS) return;
    const int v = vid[pair];
    const int e = eid[pair];
    const int c0 = lane * 4;
    const float4 xe = *(const float4*)(Xe + (size_t)e * H + c0);
    float* dst = out + (size_t)v * H + c0;
    atomicAdd(dst + 0, xe.x);
    atomicAdd(dst + 1, xe.y);
    atomicAdd(dst + 2, xe.z);
    atomicAdd(dst + 3, xe.w);
    if (lane == 0) atomicAdd(vcnt + v, 1.0f);
}

// ------------------------------------------------------- divide + ReLU
__global__ void final_relu(float* __restrict__ out, const float* __restrict__ cnt,
                           long total) {
    long idx = (long)blockIdx.x * blockDim.x + threadIdx.x;
    if (idx < total) {
        long r = idx >> 7;
        float v = out[idx] / fmaxf(cnt[r], 1.0f);
        out[idx] = fmaxf(v, 0.0f);
    }
}

extern "C" void kernel_launch(void* const* d_in, const int* in_sizes, int n_in,
                              void* d_out, int out_size, void* d_ws, size_t ws_size,
                              hipStream_t stream) {
    const float* X     = (const float*)d_in[0];
    const int*   vid   = (const int*)  d_in[1];
    const int*   eid   = (const int*)  d_in[2];
    const float* W     = (const float*)d_in[3];
    const float* bias  = (const float*)d_in[4];
    const float* gamma = (const float*)d_in[5];
    const float* beta  = (const float*)d_in[6];
    float* out = (float*)d_out;

    // workspace layout (floats)
    float* ws     = (float*)d_ws;
    float* h      = ws;                          // N_V*H
    float* Xe     = h    + (size_t)N_V * H;      // N_E*H
    float* ecnt   = Xe   + (size_t)N_E * H;      // N_E
    float* vcnt   = ecnt + N_E;                  // N_V
    float* csum   = vcnt + N_V;                  // H
    float* csumsq = csum + H;                    // H
    float* cscale = csumsq + H;                  // H
    float* cshift = cscale + H;                  // H

    // zero the accumulators (Xe .. csumsq are contiguous) and the output
    const long zlen = (long)N_E * H + N_E + N_V + 2 * H;
    zero_f32<<<2048, 256, 0, stream>>>(Xe, zlen);
    zero_f32<<<2048, 256, 0, stream>>>(out, (long)N_V * H);

    // GEMM + BN statistics
    gemm_bn_stats<<<N_V / 16, 256, 0, stream>>>(X, W, bias, h, csum, csumsq);
    bn_finalize<<<1, 128, 0, stream>>>(csum, csumsq, gamma, beta, cscale, cshift);

    // vertex -> hyperedge mean
    scatter_v2e<<<(N_PAIRS + 7) / 8, 256, 0, stream>>>(h, vid, eid, cscale, cshift,
                                                       Xe, ecnt);
    seg_div<<<((long)N_E * H + 255) / 256, 256, 0, stream>>>(Xe, ecnt, (long)N_E * H);

    // hyperedge -> vertex mean, then ReLU
    scatter_e2v<<<(N_PAIRS + 7) / 8, 256, 0, stream>>>(Xe, vid, eid, out, vcnt);
    final_relu<<<((long)N_V * H + 255) / 256, 256, 0, stream>>>(out, vcnt,
                                                                (long)N_V * H);
}